// MultiHeadGraphAttention_7413113553205
// MI455X (gfx1250) — compile-verified
//
#include <hip/hip_runtime.h>

#define N_NODES  50000
#define N_EDGES  800000
#define DIM      256
#define HEADS    4
#define HEAD_DIM 64
#define MTILE    80              // 50000 = 625 * 80; 5 row-tiles of 16 per block
#define MBLOCKS  (N_NODES / MTILE)

typedef __attribute__((ext_vector_type(2))) float v2f;
typedef __attribute__((ext_vector_type(8))) float v8f;

// ---- CDNA5 async global->LDS helpers --------------------------------------
__device__ __forceinline__ unsigned lds_off(const void* p) {
    // generic -> LDS(AS3) pointer, truncated to the 32-bit LDS byte address
    return (unsigned)(unsigned long long)(__attribute__((address_space(3))) char*)
           const_cast<void*>(p);
}

__device__ __forceinline__ void async_load_b128(unsigned lds_byte_addr, const void* gptr) {
#if defined(__AMDGCN__)
    asm volatile("global_load_async_to_lds_b128 %0, %1, off"
                 :: "v"(lds_byte_addr), "v"((unsigned long long)gptr)
                 : "memory");
#endif
}

__device__ __forceinline__ void wait_async0() {
#if __has_builtin(__builtin_amdgcn_s_wait_asynccnt)
    __builtin_amdgcn_s_wait_asynccnt(0);
#elif defined(__AMDGCN__)
    asm volatile("s_wait_asynccnt 0x0" ::: "memory");
#endif
}

// Order-preserving float<->uint mapping so float-max can use u32 atomicMax.
__device__ __forceinline__ unsigned enc_f32(float f) {
    unsigned u = __float_as_uint(f);
    return (u & 0x80000000u) ? ~u : (u | 0x80000000u);
}
__device__ __forceinline__ float dec_f32(unsigned u) {
    unsigned b = (u & 0x80000000u) ? (u & 0x7FFFFFFFu) : ~u;
    return __uint_as_float(b);
}

// ---------------------------------------------------------------------------
// Zero the scatter-softmax accumulators (segmax encoded as 0 == below -inf).
// ---------------------------------------------------------------------------
__global__ void init_buffers(unsigned* __restrict__ segmax, float* __restrict__ denom) {
    int gid = blockIdx.x * blockDim.x + threadIdx.x;
    if (gid < N_NODES * HEADS) {
        segmax[gid] = 0u;
        denom[gid]  = 0.0f;
    }
}

// ---------------------------------------------------------------------------
// Fused Q/K/V/Skip projection: out = x @ B (+bias), B[d, h*64+k] = W[h,d,k].
// blockIdx.y selects the weight set.  Block = 80 rows x 256 cols, 8 waves.
// x-tile is async-DMA'd to LDS; each wave holds 5 row-tiles x 2 col-tiles so
// every B fragment feeds 5 v_wmma_f32_16x16x4_f32 ops.
// ---------------------------------------------------------------------------
__global__ __launch_bounds__(256) void proj_gemm(
    const float* __restrict__ x,
    const float* __restrict__ Wq, const float* __restrict__ bq,
    const float* __restrict__ Wk, const float* __restrict__ bk,
    const float* __restrict__ Wv, const float* __restrict__ bv,
    const float* __restrict__ Ws, const float* __restrict__ bs,
    float* __restrict__ q, float* __restrict__ k,
    float* __restrict__ v, float* __restrict__ agg)
{
    __shared__ float s_tile[MTILE][260];     // 83.2 KB, stride 260 -> no bank conflicts

    const float* W; const float* bias; float* out;
    switch (blockIdx.y) {
        case 0:  W = Wq; bias = bq; out = q;   break;
        case 1:  W = Wk; bias = bk; out = k;   break;
        case 2:  W = Wv; bias = bv; out = v;   break;
        default: W = Ws; bias = bs; out = agg; break;
    }
    const int tid = threadIdx.x;
    const int m0  = blockIdx.x * MTILE;

    // Async-stage the 80x256 x tile (16B chunks, coalesced).
    #pragma unroll
    for (int i = 0; i < 20; ++i) {
        const int c   = i * 256 + tid;       // 0..5119 float4 chunks
        const int row = c >> 6;              // 64 chunks per row
        const int col = (c & 63) << 2;
        async_load_b128(lds_off(&s_tile[row][col]),
                        x + (size_t)(m0 + row) * DIM + col);
    }
    wait_async0();
    __syncthreads();

    const int lane = tid & 31;
    const int wave = tid >> 5;
    const int m  = lane & 15;                // A row within tile
    const int kh = lane >> 4;                // K pair {0,1} vs {2,3}
    const int n0 = lane & 15;                // B/D column within tile
    const int c0 = wave * 32;                // col tiles c0 and c0+16

    // W[h,d,kc] flat = h*16384 + d*64 + kc ; col c -> h=c>>6, kc=c&63
    const int h0 = c0 >> 6;          const int col0 = (c0 & 63) + n0;
    const int h1 = (c0 + 16) >> 6;   const int col1 = ((c0 + 16) & 63) + n0;
    const float* Wp0 = W + (size_t)h0 * (DIM * HEAD_DIM) + col0;
    const float* Wp1 = W + (size_t)h1 * (DIM * HEAD_DIM) + col1;

    v8f acc[5][2] = {};
    #pragma unroll 2
    for (int kk = 0; kk < DIM; kk += 4) {
        const int d = kk + 2 * kh;
        v2f b0, b1;
        b0.x = Wp0[(size_t)d * HEAD_DIM];   b0.y = Wp0[(size_t)(d + 1) * HEAD_DIM];
        b1.x = Wp1[(size_t)d * HEAD_DIM];   b1.y = Wp1[(size_t)(d + 1) * HEAD_DIM];
        #pragma unroll
        for (int rt = 0; rt < 5; ++rt) {
            v2f a = *(const v2f*)&s_tile[rt * 16 + m][d];   // ds_load_b64
            acc[rt][0] = __builtin_amdgcn_wmma_f32_16x16x4_f32(
                false, a, false, b0, (short)0, acc[rt][0], false, false);
            acc[rt][1] = __builtin_amdgcn_wmma_f32_16x16x4_f32(
                false, a, false, b1, (short)0, acc[rt][1], false, false);
        }
    }

    const int rbase = 8 * kh;
    const int cA = c0 + n0;
    const int cB = c0 + 16 + n0;
    const float biasA = bias[cA], biasB = bias[cB];
    #pragma unroll
    for (int rt = 0; rt < 5; ++rt) {
        #pragma unroll
        for (int r = 0; r < 8; ++r) {
            const int row = m0 + rt * 16 + rbase + r;
            out[(size_t)row * DIM + cA] = acc[rt][0][r] + biasA;
            out[(size_t)row * DIM + cB] = acc[rt][1][r] + biasB;
        }
    }
}

// ---------------------------------------------------------------------------
// Per-edge attention logits + segment-max.  One 32-lane wave per edge; lane t
// covers dims [8t, 8t+8) -> head = t/8.  q/k gathers resolve in the 192MB L2.
// ---------------------------------------------------------------------------
__global__ __launch_bounds__(256) void edge_scores(
    const float* __restrict__ q, const float* __restrict__ k,
    const int* __restrict__ ei, float* __restrict__ scores,
    unsigned* __restrict__ segmax)
{
    const int lane = threadIdx.x & 31;
    const int e = blockIdx.x * 8 + (threadIdx.x >> 5);
    if (e >= N_EDGES) return;
    const int src = ei[e];
    const int dst = ei[N_EDGES + e];
    const int d0 = lane * 8;
    const float4* qp = (const float4*)(q + (size_t)dst * DIM + d0);
    const float4* kp = (const float4*)(k + (size_t)src * DIM + d0);
    float4 qa = qp[0], qb = qp[1];
    float4 ka = kp[0], kb = kp[1];
    float p = qa.x * ka.x + qa.y * ka.y + qa.z * ka.z + qa.w * ka.w
            + qb.x * kb.x + qb.y * kb.y + qb.z * kb.z + qb.w * kb.w;
    p += __shfl_xor(p, 1);
    p += __shfl_xor(p, 2);
    p += __shfl_xor(p, 4);                  // groups of 8 lanes -> per-head sum
    if ((lane & 7) == 0) {
        const int h = lane >> 3;
        const float s = p * 0.125f;         // 1/sqrt(64)
        scores[(size_t)e * HEADS + h] = s;
        atomicMax(segmax + (size_t)dst * HEADS + h, enc_f32(s));
    }
}

// ---------------------------------------------------------------------------
// ex = exp(score - segmax[dst]); denom[dst] += ex.  One thread per (edge,head).
// ---------------------------------------------------------------------------
__global__ __launch_bounds__(256) void edge_softmax_num(
    const int* __restrict__ ei, float* __restrict__ scores,
    const unsigned* __restrict__ segmax, float* __restrict__ denom)
{
    const long long gid = (long long)blockIdx.x * blockDim.x + threadIdx.x;
    if (gid >= (long long)N_EDGES * HEADS) return;
    const int e = (int)(gid >> 2);
    const int h = (int)(gid & 3);
    const int dst = ei[N_EDGES + e];
    const float sm = dec_f32(segmax[(size_t)dst * HEADS + h]);
    const float ex = expf(scores[gid] - sm);
    scores[gid] = ex;                       // overwrite with numerator
    atomicAdd(denom + (size_t)dst * HEADS + h, ex);
}

// ---------------------------------------------------------------------------
// agg[dst] += alpha * v[src].  Wave per edge, lane covers 8 dims of its head.
// agg was pre-initialized with the skip projection by proj_gemm.
// ---------------------------------------------------------------------------
__global__ __launch_bounds__(256) void edge_scatter_v(
    const int* __restrict__ ei, const float* __restrict__ scores,
    const float* __restrict__ denom, const float* __restrict__ v,
    float* __restrict__ agg)
{
    const int lane = threadIdx.x & 31;
    const int e = blockIdx.x * 8 + (threadIdx.x >> 5);
    if (e >= N_EDGES) return;
    const int src = ei[e];
    const int dst = ei[N_EDGES + e];
    const int h = lane >> 3;
    const float alpha = scores[(size_t)e * HEADS + h] /
                        (denom[(size_t)dst * HEADS + h] + 1e-16f);
    const int d0 = lane * 8;
    const float4* vp = (const float4*)(v + (size_t)src * DIM + d0);
    float4 va = vp[0], vb = vp[1];
    float* ap = agg + (size_t)dst * DIM + d0;
    atomicAdd(ap + 0, alpha * va.x);
    atomicAdd(ap + 1, alpha * va.y);
    atomicAdd(ap + 2, alpha * va.z);
    atomicAdd(ap + 3, alpha * va.w);
    atomicAdd(ap + 4, alpha * vb.x);
    atomicAdd(ap + 5, alpha * vb.y);
    atomicAdd(ap + 6, alpha * vb.z);
    atomicAdd(ap + 7, alpha * vb.w);
}

// ---------------------------------------------------------------------------
// out = LN( multi @ Wout + bout + x ).  80x256 tile per block.  multi is
// async-staged to LDS; the residual x tile is async-prefetched into a second
// LDS buffer *during* the WMMA K-loop (copy/compute overlap), waited on with
// s_wait_asynccnt before the LayerNorm epilogue.
// ---------------------------------------------------------------------------
__global__ __launch_bounds__(256) void out_gemm_ln(
    const float* __restrict__ agg, const float* __restrict__ x,
    const float* __restrict__ Wout, const float* __restrict__ bout,
    const float* __restrict__ ln_g, const float* __restrict__ ln_b,
    float* __restrict__ out)
{
    __shared__ float s_tile[MTILE][260];    // multi, then h = out+bout+x
    __shared__ float s_x[MTILE][260];       // residual tile (async prefetch)
    __shared__ float s_sum[16][17];
    __shared__ float s_sq[16][17];
    __shared__ float s_mu[16];
    __shared__ float s_rs[16];

    const int tid = threadIdx.x;
    const int m0 = blockIdx.x * MTILE;

    // Stage multi = agg (skip + attention already accumulated there).
    #pragma unroll
    for (int i = 0; i < 20; ++i) {
        const int c   = i * 256 + tid;
        const int row = c >> 6;
        const int col = (c & 63) << 2;
        async_load_b128(lds_off(&s_tile[row][col]),
                        agg + (size_t)(m0 + row) * DIM + col);
    }
    wait_async0();
    __syncthreads();

    // Kick off the residual x tile; consumed only after the K-loop.
    #pragma unroll
    for (int i = 0; i < 20; ++i) {
        const int c   = i * 256 + tid;
        const int row = c >> 6;
        const int col = (c & 63) << 2;
        async_load_b128(lds_off(&s_x[row][col]),
                        x + (size_t)(m0 + row) * DIM + col);
    }

    const int lane = tid & 31;
    const int wave = tid >> 5;
    const int m  = lane & 15;
    const int kh = lane >> 4;
    const int n0 = lane & 15;
    const int c0 = wave * 32;

    v8f acc[5][2] = {};
    #pragma unroll 2
    for (int kk = 0; kk < DIM; kk += 4) {
        const int d = kk + 2 * kh;
        v2f b0, b1;
        b0.x = Wout[(size_t)d * DIM + c0 + n0];
        b0.y = Wout[(size_t)(d + 1) * DIM + c0 + n0];
        b1.x = Wout[(size_t)d * DIM + c0 + 16 + n0];
        b1.y = Wout[(size_t)(d + 1) * DIM + c0 + 16 + n0];
        #pragma unroll
        for (int rt = 0; rt < 5; ++rt) {
            v2f a = *(const v2f*)&s_tile[rt * 16 + m][d];
            acc[rt][0] = __builtin_amdgcn_wmma_f32_16x16x4_f32(
                false, a, false, b0, (short)0, acc[rt][0], false, false);
            acc[rt][1] = __builtin_amdgcn_wmma_f32_16x16x4_f32(
                false, a, false, b1, (short)0, acc[rt][1], false, false);
        }
    }
    wait_async0();       // x tile landed
    __syncthreads();     // all A reads done; s_tile reusable

    // h = out + bout + x  -> back into s_tile
    const int rbase = 8 * kh;
    const int cA = c0 + n0;
    const int cB = c0 + 16 + n0;
    const float bA = bout[cA], bB = bout[cB];
    #pragma unroll
    for (int rt = 0; rt < 5; ++rt) {
        #pragma unroll
        for (int r = 0; r < 8; ++r) {
            const int row = rt * 16 + rbase + r;
            s_tile[row][cA] = acc[rt][0][r] + bA + s_x[row][cA];
            s_tile[row][cB] = acc[rt][1][r] + bB + s_x[row][cB];
        }
    }
    __syncthreads();

    // Row-wise LayerNorm, 16 rows per pass (16 threads per row).
    for (int p = 0; p < 5; ++p) {
        const int row16 = tid >> 4, seg = tid & 15;
        const int row = p * 16 + row16;
        float s = 0.f, ss = 0.f;
        #pragma unroll
        for (int j = 0; j < 16; ++j) {
            const float hv = s_tile[row][seg * 16 + j];
            s += hv; ss += hv * hv;
        }
        s_sum[row16][seg] = s; s_sq[row16][seg] = ss;
        __syncthreads();
        if (tid < 16) {
            float t = 0.f, tt = 0.f;
            #pragma unroll
            for (int j = 0; j < 16; ++j) { t += s_sum[tid][j]; tt += s_sq[tid][j]; }
            const float mu  = t * (1.0f / 256.0f);
            const float var = tt * (1.0f / 256.0f) - mu * mu;
            s_mu[tid] = mu;
            s_rs[tid] = rsqrtf(var + 1e-5f);
        }
        __syncthreads();
        #pragma unroll
        for (int i = 0; i < 16; ++i) {
            const int idx = i * 256 + tid;
            const int r16 = idx >> 8, col = idx & 255;
            const float hv = s_tile[p * 16 + r16][col];
            out[(size_t)(m0 + p * 16 + r16) * DIM + col] =
                (hv - s_mu[r16]) * s_rs[r16] * ln_g[col] + ln_b[col];
        }
        __syncthreads();   // s_sum/s_sq reused next pass
    }
}

// ---------------------------------------------------------------------------
extern "C" void kernel_launch(void* const* d_in, const int* in_sizes, int n_in,
                              void* d_out, int out_size, void* d_ws, size_t ws_size,
                              hipStream_t stream)
{
    (void)in_sizes; (void)n_in; (void)out_size; (void)ws_size;
    const float* x    = (const float*)d_in[0];
    const int*   ei   = (const int*)d_in[1];
    const float* Wq   = (const float*)d_in[2];
    const float* bq   = (const float*)d_in[3];
    const float* Wk   = (const float*)d_in[4];
    const float* bk   = (const float*)d_in[5];
    const float* Wv   = (const float*)d_in[6];
    const float* bv   = (const float*)d_in[7];
    const float* Ws   = (const float*)d_in[8];
    const float* bs   = (const float*)d_in[9];
    const float* Wout = (const float*)d_in[10];
    const float* bout = (const float*)d_in[11];
    const float* lg   = (const float*)d_in[12];
    const float* lb   = (const float*)d_in[13];
    float* out = (float*)d_out;

    float* ws = (float*)d_ws;
    const size_t NF = (size_t)N_NODES * DIM;
    float*    q   = ws;
    float*    k   = ws + NF;
    float*    v   = ws + 2 * NF;
    float*    agg = ws + 3 * NF;                 // skip proj + attention agg
    float*    sc  = ws + 4 * NF;                 // [E, H] scores -> numerators
    unsigned* sm  = (unsigned*)(ws + 4 * NF + (size_t)N_EDGES * HEADS);
    float*    den = (float*)(sm + (size_t)N_NODES * HEADS);

    init_buffers<<<(N_NODES * HEADS + 255) / 256, 256, 0, stream>>>(sm, den);
    proj_gemm<<<dim3(MBLOCKS, 4), 256, 0, stream>>>(
        x, Wq, bq, Wk, bk, Wv, bv, Ws, bs, q, k, v, agg);
    edge_scores<<<(N_EDGES + 7) / 8, 256, 0, stream>>>(q, k, ei, sc, sm);
    edge_softmax_num<<<(int)(((long long)N_EDGES * HEADS + 255) / 256), 256, 0, stream>>>(
        ei, sc, sm, den);
    edge_scatter_v<<<(N_EDGES + 7) / 8, 256, 0, stream>>>(ei, sc, den, v, agg);
    out_gemm_ln<<<MBLOCKS, 256, 0, stream>>>(agg, x, Wout, bout, lg, lb, out);
}